// EnhancedGAT_80487687127284
// MI455X (gfx1250) — compile-verified
//
#include <hip/hip_runtime.h>
#include <math.h>

// ---------------------------------------------------------------------------
// EnhancedGAT forward on gfx1250 (MI455X), fp32 end-to-end.
//   GEMMs: V_WMMA_F32_16X16X4_F32 (wave32 WMMA), A-tile staged into LDS by
//   the Tensor Data Mover (tensor_load_to_lds + s_wait_tensorcnt), one B
//   fragment feeds MT WMMAs.
//   Edge softmax: 3 scatter passes (atomic max via monotonic uint encoding,
//   atomic add for denom, float4-vectorized atomic-add aggregation). All
//   attention side arrays (<1 MB) and h matrices (<26 MB) are L2-resident.
// Workspace: N*340 floats (~68 MB for N=50000), layout below.
// ---------------------------------------------------------------------------

typedef __attribute__((ext_vector_type(2))) float    v2f;
typedef __attribute__((ext_vector_type(8))) float    v8f;
typedef __attribute__((ext_vector_type(4))) unsigned u32x4;
typedef __attribute__((ext_vector_type(4))) int      i32x4;
typedef __attribute__((ext_vector_type(8))) int      i32x8;

#define NEG_SLOPE 0.2f

// ---------------- monotonic float<->uint encoding for atomic max ------------
__device__ __forceinline__ unsigned fenc(float f) {
  unsigned u = __float_as_uint(f);
  return (u & 0x80000000u) ? ~u : (u | 0x80000000u);
}
__device__ __forceinline__ float fdec(unsigned u) {
  unsigned b = (u & 0x80000000u) ? (u ^ 0x80000000u) : ~u;
  return __uint_as_float(b);
}

// ---------------- zero fill -------------------------------------------------
__global__ void fill_zero_kernel(float* __restrict__ p, size_t n) {
  size_t i = (size_t)blockIdx.x * blockDim.x + threadIdx.x;
  if (i < n) p[i] = 0.0f;
}

// ---------------- fp32 WMMA GEMM: C[M,NCOL] = A[M,K] @ B[K,NCOL] ------------
// Block covers ROWS = 16*MT rows; wave w computes cols [16w,16w+16) for all
// MT row tiles, so each B fragment load is reused by MT WMMAs.
// A-tile staging: one TDM tensor_load_to_lds per block (issued by wave 0,
// EXEC-independent, tracked by TENSORcnt). TDM pad fields insert 4 DWORDs
// after each K-DWORD row -> LDS row stride K+4. Stride mod 64 == 4 makes the
// f32 A-fragment read pattern {lane<16: K+0,K+1 ; lane>=16: K+2,K+3} hit all
// 64 banks conflict-free. Rows past M are zero-filled by the TDM OOB clamp
// (tensor_dim1 = M - rowBase).
template <int K, int NCOL, int NWAVE, int MT>
__global__ __launch_bounds__(NWAVE * 32)
void gemm_f32_wmma_kernel(const float* __restrict__ A, const float* __restrict__ B,
                          float* __restrict__ C, int M) {
  constexpr int LDSS = K + 4;
  constexpr int ROWS = 16 * MT;
  __shared__ float aTile[ROWS * LDSS];
  const int tid = threadIdx.x;
  const int rowBase = blockIdx.x * ROWS;
  const bool full = (rowBase + ROWS) <= M;

  if (tid == 0) {  // wave-uniform branch: only wave 0 issues the TDM op
    unsigned ldsOff = (unsigned)(size_t)aTile;  // flat->LDS: addr[31:0]
    unsigned long long ga =
        (unsigned long long)(size_t)(A + (size_t)rowBase * K);
    // ---- D# group 0: count=1 | lds_addr | global_addr | type=2 ----
    u32x4 g0;
    g0.x = 1u;
    g0.y = ldsOff;
    g0.z = (unsigned)ga;
    g0.w = (unsigned)(ga >> 32) | (2u << 30);
    // ---- D# group 1 ----
    // dword0: wg_mask=0 | data_size=2(4B)<<16 | pad_en<<20 | pad_ivl<<22
    //         | pad_amt<<25   (pad: one row of K DWORDs, then +4 DWORDs)
    constexpr unsigned padIvl = (K == 256) ? 7u : 6u;  // 256 / 128 DWORDs
    constexpr unsigned padAmt = 3u;                    // 4 DWORDs
    unsigned td0 = (unsigned)K;                 // tensor dim0 (elements/row)
    unsigned td1 = (unsigned)(M - rowBase);     // remaining rows (OOB -> 0)
    i32x8 g1;
    g1[0] = (int)((2u << 16) | (1u << 20) | (padIvl << 22) | (padAmt << 25));
    g1[1] = (int)((td0 & 0xFFFFu) << 16);               // tensor_dim0[15:0]
    g1[2] = (int)((td0 >> 16) | ((td1 & 0xFFFFu) << 16));
    g1[3] = (int)((td1 >> 16) | ((unsigned)K << 16));   // tile_dim0 = K
    g1[4] = (int)ROWS;                                  // tile_dim1, tile_dim2=0
    g1[5] = (int)K;                                     // dim0 stride = K
    g1[6] = 0;
    g1[7] = 0;
    i32x4 gz4 = {0, 0, 0, 0};                // groups 2/3 unused (2D tensor)
    i32x8 gz8 = {0, 0, 0, 0, 0, 0, 0, 0};    // trailing group (unused)
    __builtin_amdgcn_tensor_load_to_lds(g0, g1, gz4, gz4, gz8, 0);
    __builtin_amdgcn_s_wait_tensorcnt(0);
  }
  __syncthreads();

  const int wave = tid >> 5;
  const int lane = tid & 31;
  const int half = lane >> 4;   // 0: K+0/K+1 ; 1: K+2/K+3  (f32 frag split)
  const int l16  = lane & 15;
  const int col0 = wave * 16;

  v8f acc[MT];
#pragma unroll
  for (int m = 0; m < MT; ++m) acc[m] = (v8f){};

  const float* bCol = &B[(size_t)(half * 2) * NCOL + col0 + l16];
  const float* aBase = &aTile[l16 * LDSS + half * 2];

  for (int k0 = 0; k0 < K; k0 += 4) {
    v2f b;
    b.x = bCol[(size_t)k0 * NCOL];
    b.y = bCol[(size_t)(k0 + 1) * NCOL];
#pragma unroll
    for (int m = 0; m < MT; ++m) {
      const float* aRow = aBase + m * 16 * LDSS + k0;
      v2f a;
      a.x = aRow[0];
      a.y = aRow[1];
      acc[m] = __builtin_amdgcn_wmma_f32_16x16x4_f32(
          /*neg_a=*/false, a, /*neg_b=*/false, b,
          /*c_mod=*/(short)0, acc[m], /*reuse_a=*/false, /*reuse_b=*/false);
    }
  }

  // C/D layout: VGPR r -> rows {r, r+8}; lanes split at 16 for N
  if (full) {
#pragma unroll
    for (int m = 0; m < MT; ++m)
#pragma unroll
      for (int r = 0; r < 8; ++r) {
        int row = rowBase + m * 16 + r + half * 8;
        C[(size_t)row * NCOL + col0 + l16] = acc[m][r];
      }
  } else {
#pragma unroll
    for (int m = 0; m < MT; ++m)
#pragma unroll
      for (int r = 0; r < 8; ++r) {
        int row = rowBase + m * 16 + r + half * 8;
        if (row < M) C[(size_t)row * NCOL + col0 + l16] = acc[m][r];
      }
  }
}

// ---------------- per-node attention logits: al[n,h] = <h[n,h,:], att[h,:]> -
template <int H, int C>
__global__ void attn_logits_kernel(const float* __restrict__ h,
                                   const float* __restrict__ att_s,
                                   const float* __restrict__ att_d,
                                   float* __restrict__ al_s,
                                   float* __restrict__ al_d, int n) {
  size_t id = (size_t)blockIdx.x * blockDim.x + threadIdx.x;
  if (id >= (size_t)n * H) return;
  int node = (int)(id / H);
  int hh = (int)(id % H);
  const float* hr = h + (size_t)node * H * C + hh * C;
  float s = 0.0f, d = 0.0f;
#pragma unroll 8
  for (int c = 0; c < C; ++c) {
    float v = hr[c];
    s += v * att_s[hh * C + c];
    d += v * att_d[hh * C + c];
  }
  al_s[id] = s;
  al_d[id] = d;
}

// edge endpoints + leaky-relu'd logit, shared by all three passes
template <int H>
__device__ __forceinline__ float edge_logit(const int* __restrict__ es,
                                            const int* __restrict__ ed, int E,
                                            size_t eid, int hh,
                                            const float* __restrict__ al_s,
                                            const float* __restrict__ al_d,
                                            int& sOut, int& dOut) {
  int s, d;
  if (eid < (size_t)E) {
    s = es[eid];
    d = ed[eid];
  } else {
    s = d = (int)(eid - E);  // self-loop
  }
  sOut = s;
  dOut = d;
  float e = al_s[(size_t)s * H + hh] + al_d[(size_t)d * H + hh];
  return (e > 0.0f) ? e : NEG_SLOPE * e;
}

// ---------------- pass 1: segment max over dst ------------------------------
template <int H>
__global__ void edge_max_kernel(const int* __restrict__ es, const int* __restrict__ ed,
                                int E, int N,
                                const float* __restrict__ al_s,
                                const float* __restrict__ al_d,
                                unsigned* __restrict__ emax) {
  size_t id = (size_t)blockIdx.x * blockDim.x + threadIdx.x;
  size_t total = (size_t)(E + N) * H;
  if (id >= total) return;
  size_t eid = id / H;
  int hh = (int)(id % H);
  int s, d;
  float e = edge_logit<H>(es, ed, E, eid, hh, al_s, al_d, s, d);
  atomicMax(&emax[(size_t)d * H + hh], fenc(e));
}

// ---------------- pass 2: segment sum of exp(e - max) -----------------------
template <int H>
__global__ void edge_sum_kernel(const int* __restrict__ es, const int* __restrict__ ed,
                                int E, int N,
                                const float* __restrict__ al_s,
                                const float* __restrict__ al_d,
                                const unsigned* __restrict__ emax,
                                float* __restrict__ denom) {
  size_t id = (size_t)blockIdx.x * blockDim.x + threadIdx.x;
  size_t total = (size_t)(E + N) * H;
  if (id >= total) return;
  size_t eid = id / H;
  int hh = (int)(id % H);
  int s, d;
  float e = edge_logit<H>(es, ed, E, eid, hh, al_s, al_d, s, d);
  float m = fdec(emax[(size_t)d * H + hh]);
  atomicAdd(&denom[(size_t)d * H + hh], expf(e - m));
}

// ---------------- pass 3: out[dst] += alpha * h[src], float4 chunks ---------
template <int H, int C>
__global__ void edge_agg_kernel(const int* __restrict__ es, const int* __restrict__ ed,
                                int E, int N,
                                const float* __restrict__ al_s,
                                const float* __restrict__ al_d,
                                const unsigned* __restrict__ emax,
                                const float* __restrict__ denom,
                                const float* __restrict__ h,
                                float* __restrict__ out) {
  constexpr int HC4 = H * C / 4;  // 4-float chunks per edge
  size_t id = (size_t)blockIdx.x * blockDim.x + threadIdx.x;
  size_t total = (size_t)(E + N) * HC4;
  if (id >= total) return;
  size_t eid = id / HC4;
  int chunk = (int)(id % HC4);
  int feat = chunk * 4;   // C % 4 == 0 -> chunk never straddles heads
  int hh = feat / C;
  int s, d;
  float e = edge_logit<H>(es, ed, E, eid, hh, al_s, al_d, s, d);
  float m = fdec(emax[(size_t)d * H + hh]);
  float alpha = expf(e - m) / denom[(size_t)d * H + hh];
  const float4 hv = *(const float4*)&h[(size_t)s * H * C + feat];
  float* o = &out[(size_t)d * H * C + feat];
  atomicAdd(o + 0, hv.x * alpha);
  atomicAdd(o + 1, hv.y * alpha);
  atomicAdd(o + 2, hv.z * alpha);
  atomicAdd(o + 3, hv.w * alpha);
}

// ---------------- bias (+ optional ELU), in place ---------------------------
template <int MASK, bool ELU>
__global__ void bias_act_kernel(float* __restrict__ h, const float* __restrict__ b,
                                size_t n) {
  size_t i = (size_t)blockIdx.x * blockDim.x + threadIdx.x;
  if (i >= n) return;
  float v = h[i] + b[i & (size_t)MASK];
  if (ELU) v = (v > 0.0f) ? v : (expf(v) - 1.0f);
  h[i] = v;
}

// ---------------------------------------------------------------------------
static inline unsigned gridFor(size_t n, int block) {
  return (unsigned)((n + block - 1) / block);
}

extern "C" void kernel_launch(void* const* d_in, const int* in_sizes, int n_in,
                              void* d_out, int out_size, void* d_ws, size_t ws_size,
                              hipStream_t stream) {
  const float* x      = (const float*)d_in[0];
  const int*   ei     = (const int*)d_in[1];
  const float* W1     = (const float*)d_in[2];
  const float* att_s1 = (const float*)d_in[3];
  const float* att_d1 = (const float*)d_in[4];
  const float* b1     = (const float*)d_in[5];
  const float* W2     = (const float*)d_in[6];
  const float* att_s2 = (const float*)d_in[7];
  const float* att_d2 = (const float*)d_in[8];
  const float* b2     = (const float*)d_in[9];
  float* out = (float*)d_out;

  const int N = in_sizes[0] / 256;   // F_IN = 256
  const int E = in_sizes[1] / 2;     // edge_index is [2, E]
  const int* es = ei;
  const int* ed = ei + E;
  const size_t ET = (size_t)E + (size_t)N;  // with self-loops

  // workspace carve-up
  float* ws     = (float*)d_ws;
  float* h1     = ws;                       // N*128
  float* agg1   = h1 + (size_t)N * 128;     // N*128 (becomes h_elu in place)
  float* h2     = agg1 + (size_t)N * 128;   // N*64
  float* al_s1  = h2 + (size_t)N * 64;      // N*4
  float* al_d1  = al_s1 + (size_t)N * 4;    // N*4
  float* al_s2  = al_d1 + (size_t)N * 4;    // N
  float* al_d2  = al_s2 + (size_t)N;        // N
  float* zreg   = al_d2 + (size_t)N;        // zero-init region: N*10 floats
  float*    denom1 = zreg;                  // N*4
  unsigned* emax1  = (unsigned*)(denom1 + (size_t)N * 4); // N*4
  float*    denom2 = (float*)(emax1 + (size_t)N * 4);     // N
  unsigned* emax2  = (unsigned*)(denom2 + (size_t)N);     // N

  const int BLK = 256;

  // --- init accumulators (deterministic each call) ---
  fill_zero_kernel<<<gridFor((size_t)N * 128, BLK), BLK, 0, stream>>>(agg1, (size_t)N * 128);
  fill_zero_kernel<<<gridFor((size_t)N * 64,  BLK), BLK, 0, stream>>>(out,  (size_t)N * 64);
  fill_zero_kernel<<<gridFor((size_t)N * 10,  BLK), BLK, 0, stream>>>(zreg, (size_t)N * 10);

  // ================= layer 1: F_IN=256 -> H=4, C=32 (concat) ===============
  {
    constexpr int MT = 4;
    dim3 g(gridFor((size_t)N, 16 * MT));
    gemm_f32_wmma_kernel<256, 128, 8, MT><<<g, 256, 0, stream>>>(x, W1, h1, N);
  }

  attn_logits_kernel<4, 32><<<gridFor((size_t)N * 4, BLK), BLK, 0, stream>>>(
      h1, att_s1, att_d1, al_s1, al_d1, N);

  edge_max_kernel<4><<<gridFor(ET * 4, BLK), BLK, 0, stream>>>(
      es, ed, E, N, al_s1, al_d1, emax1);
  edge_sum_kernel<4><<<gridFor(ET * 4, BLK), BLK, 0, stream>>>(
      es, ed, E, N, al_s1, al_d1, emax1, denom1);
  edge_agg_kernel<4, 32><<<gridFor(ET * 32, BLK), BLK, 0, stream>>>(
      es, ed, E, N, al_s1, al_d1, emax1, denom1, h1, agg1);

  bias_act_kernel<127, true><<<gridFor((size_t)N * 128, BLK), BLK, 0, stream>>>(
      agg1, b1, (size_t)N * 128);

  // ================= layer 2: 128 -> OUT=64, 1 head, no concat =============
  {
    constexpr int MT = 4;
    dim3 g(gridFor((size_t)N, 16 * MT));
    gemm_f32_wmma_kernel<128, 64, 4, MT><<<g, 128, 0, stream>>>(agg1, W2, h2, N);
  }

  attn_logits_kernel<1, 64><<<gridFor((size_t)N, BLK), BLK, 0, stream>>>(
      h2, att_s2, att_d2, al_s2, al_d2, N);

  edge_max_kernel<1><<<gridFor(ET, BLK), BLK, 0, stream>>>(
      es, ed, E, N, al_s2, al_d2, emax2);
  edge_sum_kernel<1><<<gridFor(ET, BLK), BLK, 0, stream>>>(
      es, ed, E, N, al_s2, al_d2, emax2, denom2);
  edge_agg_kernel<1, 64><<<gridFor(ET * 16, BLK), BLK, 0, stream>>>(
      es, ed, E, N, al_s2, al_d2, emax2, denom2, h2, out);

  bias_act_kernel<63, false><<<gridFor((size_t)N * 64, BLK), BLK, 0, stream>>>(
      out, b2, (size_t)N * 64);
}